// CondenseSFR_66735201845364
// MI455X (gfx1250) — compile-verified
//
#include <hip/hip_runtime.h>

typedef __attribute__((ext_vector_type(2))) float v2f;
typedef __attribute__((ext_vector_type(8))) float v8f;

#define EPS 1e-5f
// K-pair-interleaved LDS layout: element (c, m) at float offset (c>>1)*PR + m*2 + (c&1).
// PR = 96 floats per row-pair => row-pair stride = 24 banks*4... 96 dwords ≡ 32 mod 64,
// so half-wave pair p and p+1 hit disjoint bank halves; m*2 covers 32 even banks.
#define PR 96

// Fused: BN(inference) + ReLU + channel-shuffle(transposed) + grouped 1x1 conv + reindex matmul
// x:(32,512,32,32) f32, weight:(512,128) f32, index:(512,512) f32, out:(32,512,32,32) f32
__global__ __launch_bounds__(256) void condense_sfr_fused(
    const float* __restrict__ x,
    const float* __restrict__ gamma,
    const float* __restrict__ beta,
    const float* __restrict__ rmean,
    const float* __restrict__ rvar,
    const float* __restrict__ weight,   // [512][128]
    const float* __restrict__ index,    // [512][512] row-major [c][d]
    float* __restrict__ out)
{
    __shared__ float As[256 * PR];   // shuffled/BN'd/ReLU'd input, pair-interleaved
    __shared__ float Cs[256 * PR];   // grouped-conv output,        pair-interleaved

    const int bx  = blockIdx.x;
    const int n   = bx >> 5;          // image
    const int hw0 = (bx & 31) << 5;   // 32-wide HW tile

    const int tid  = threadIdx.x;
    const int lane = tid & 31;
    const int wave = tid >> 5;        // 0..7
    const int half = lane >> 4;       // K-half select for WMMA frags
    const int l16  = lane & 15;

    // ---------------- Stage A: BN + ReLU + shuffle into LDS ----------------
    {
        const int m    = tid & 31;    // hw within tile
        const int cblk = tid >> 5;    // 8 channels per sweep
        const size_t xbase = ((size_t)n * 512) * 1024 + hw0 + m;
        for (int cp = cblk; cp < 512; cp += 8) {
            const int j = cp >> 7;            // group of shuffled channel
            const int i = cp & 127;
            const int c = i * 4 + j;          // source channel before shuffle
            const float sc = gamma[c] * rsqrtf(rvar[c] + EPS);
            const float sh = beta[c] - rmean[c] * sc;
            const float v  = x[xbase + (size_t)c * 1024];
            As[(cp >> 1) * PR + m * 2 + (cp & 1)] = fmaxf(v * sc + sh, 0.0f);
        }
    }
    __syncthreads();

    // ---------------- Stage B: grouped 1x1 conv via WMMA f32 16x16x4 ----------------
    // conv[co][m] = sum_{ci<128} W[co][ci] * As[g*128+ci][m],  g = co>>7
    // Wave w owns co-blocks [4w, 4w+4), both 16-wide m-blocks.
    {
        const int cb_lo = wave * 4;
        for (int cb = cb_lo; cb < cb_lo + 4; ++cb) {
            const int co0 = cb << 4;
            const int g   = co0 >> 7;
            v8f acc0 = {};
            v8f acc1 = {};
            const float* __restrict__ Wrow = weight + (size_t)(co0 + l16) * 128;
            #pragma unroll 4
            for (int kb = 0; kb < 128; kb += 4) {
                const int ka = kb + half * 2;                  // even
                const v2f a  = *(const v2f*)&Wrow[ka];         // A[M=co][K..K+1], 8B load
                const int p  = ((g * 128 + ka) >> 1) * PR;
                const v2f b0 = *(const v2f*)&As[p + l16 * 2];        // {row ka, ka+1} @ m
                const v2f b1 = *(const v2f*)&As[p + (l16 + 16) * 2]; // @ m+16
                acc0 = __builtin_amdgcn_wmma_f32_16x16x4_f32(
                    false, a, false, b0, (short)0, acc0, false, false);
                acc1 = __builtin_amdgcn_wmma_f32_16x16x4_f32(
                    false, a, false, b1, (short)0, acc1, false, false);
            }
            // D layout: VGPR v -> rows co0+v (lanes 0-15) / co0+v+8 (lanes 16-31);
            // rows 2p,2p+1 at same m are adjacent floats -> paired 8B stores.
            #pragma unroll
            for (int v = 0; v < 8; v += 2) {
                const int row = co0 + v + half * 8;            // even
                v2f s0; s0.x = acc0[v]; s0.y = acc0[v + 1];
                v2f s1; s1.x = acc1[v]; s1.y = acc1[v + 1];
                *(v2f*)&Cs[(row >> 1) * PR + l16 * 2]        = s0;
                *(v2f*)&Cs[(row >> 1) * PR + (l16 + 16) * 2] = s1;
            }
        }
    }
    __syncthreads();

    // ---------------- Stage C: reindex matmul via WMMA f32 16x16x4 ----------------
    // out[d][m] = sum_c index[c][d] * Cs[c][m]   (A = index^T, K = 512)
    {
        const int db_lo = wave * 4;
        for (int db = db_lo; db < db_lo + 4; ++db) {
            const int d0 = db << 4;
            v8f acc0 = {};
            v8f acc1 = {};
            #pragma unroll 4
            for (int kb = 0; kb < 512; kb += 4) {
                const int ka = kb + half * 2;                  // even
                v2f a;                                         // A[M=d][K=c] = index[c][d]
                a.x = index[(size_t)ka * 512 + d0 + l16];
                a.y = index[(size_t)(ka + 1) * 512 + d0 + l16];
                const int p  = (ka >> 1) * PR;
                const v2f b0 = *(const v2f*)&Cs[p + l16 * 2];
                const v2f b1 = *(const v2f*)&Cs[p + (l16 + 16) * 2];
                acc0 = __builtin_amdgcn_wmma_f32_16x16x4_f32(
                    false, a, false, b0, (short)0, acc0, false, false);
                acc1 = __builtin_amdgcn_wmma_f32_16x16x4_f32(
                    false, a, false, b1, (short)0, acc1, false, false);
            }
            #pragma unroll
            for (int v = 0; v < 8; ++v) {
                const int d = d0 + v + half * 8;
                const size_t base = (((size_t)n * 512 + d) << 10) + hw0;
                out[base + l16]      = acc0[v];   // 64B coalesced per 16 lanes
                out[base + 16 + l16] = acc1[v];
            }
        }
    }
}

extern "C" void kernel_launch(void* const* d_in, const int* in_sizes, int n_in,
                              void* d_out, int out_size, void* d_ws, size_t ws_size,
                              hipStream_t stream) {
    (void)n_in; (void)out_size; (void)d_ws; (void)ws_size;
    const float* x      = (const float*)d_in[0];
    const float* gamma  = (const float*)d_in[1];
    const float* beta   = (const float*)d_in[2];
    const float* rmean  = (const float*)d_in[3];
    const float* rvar   = (const float*)d_in[4];
    const float* weight = (const float*)d_in[5];
    const float* index  = (const float*)d_in[6];
    float* out = (float*)d_out;

    const int N = in_sizes[0] / (512 * 1024);   // 32
    dim3 grid(N * 32);                          // 32 HW-tiles per image
    dim3 block(256);                            // 8 wave32s
    condense_sfr_fused<<<grid, block, 0, stream>>>(x, gamma, beta, rmean, rvar,
                                                   weight, index, out);
}